// MeanAggregator_56599079026851
// MI455X (gfx1250) — compile-verified
//
#include <hip/hip_runtime.h>
#include <hip/hip_bf16.h>
#include <stdint.h>

// MeanAggregator: out[n,:] = (1/K) * sum_k embed[idx[n,k], :]
// N=100000 nodes, K=10 neighbors, D=256 fp32.
// Roofline: 102.4MB NT output write + 51.2MB embed (L2-resident, 192MB L2) ->
// HBM floor ~7us; kernel is L2-gather bound. Strategy: clause'd async
// global->LDS gather pipeline (ASYNCcnt), double buffered per wave; scalar
// (SMEM) index fetch; non-temporal output stores to protect L2 residency.

#define KNBR 10
#define DIM 256
#define HALF_BYTES 512                 // bytes of one row-half handled by a wave
#define BUF_BYTES (KNBR * HALF_BYTES)  // 5120 B per buffer
#define WAVES_PER_BLK 8

typedef float v4f __attribute__((ext_vector_type(4)));

extern __shared__ __align__(16) unsigned char lds_raw[];

__global__ void mean_agg_async_kernel(const int* __restrict__ nidx,
                                      const float* __restrict__ embed,
                                      float* __restrict__ out,
                                      int n_nodes)
{
    const int lane = threadIdx.x & 31;
    // Force wave id scalar so node/index addressing is provably uniform ->
    // index loads become SMEM s_load (KMcnt), off the vector memory pipe.
    const int waveInBlk = __builtin_amdgcn_readfirstlane((int)(threadIdx.x >> 5));
    const int wavesBlk  = blockDim.x >> 5;
    const int gwave     = blockIdx.x * wavesBlk + waveInBlk;
    const int nwaves    = gridDim.x * wavesBlk;

    const int totalHalves = n_nodes * 2;
    const int laneByte = lane * 16;

    // Per-wave private double buffer in LDS (no cross-wave sharing -> no
    // barriers; pacing purely via ASYNCcnt / DScnt).
    unsigned char* wbase = lds_raw + (size_t)waveInBlk * (2u * BUF_BYTES);
    unsigned char* pbuf0 = wbase;
    unsigned char* pbuf1 = wbase + BUF_BYTES;
    // Per-lane 32-bit LDS addresses for the async VDST operand; the k*512
    // stride is carried by the instruction offset (added to BOTH addresses).
    const unsigned la0 = (unsigned)(uintptr_t)pbuf0 + (unsigned)laneByte;
    const unsigned la1 = (unsigned)(uintptr_t)pbuf1 + (unsigned)laneByte;

    // Issue K async B128 gathers for one half-row tile as a single clause.
    // Global VADDR is biased by -k*512 so offset:k*512 lands the data at
    // lds_base + k*512 while hitting embed_row + half_off in memory.
    auto issue_half = [&](int half_id, unsigned ldsLane) {
        const int node = half_id >> 1;
        const int hoff = (half_id & 1) * HALF_BYTES;
        const int* __restrict__ ip = nidx + node * KNBR;  // scalar -> s_load
        int g[KNBR];
#pragma unroll
        for (int k = 0; k < KNBR; ++k)
            g[k] = (int)((unsigned)ip[k] * (DIM * 4u)) + hoff + laneByte
                   - k * HALF_BYTES;
        asm volatile(
            "s_clause 0x9\n\t"
            "global_load_async_to_lds_b128 %0, %1, %11 offset:0\n\t"
            "global_load_async_to_lds_b128 %0, %2, %11 offset:512\n\t"
            "global_load_async_to_lds_b128 %0, %3, %11 offset:1024\n\t"
            "global_load_async_to_lds_b128 %0, %4, %11 offset:1536\n\t"
            "global_load_async_to_lds_b128 %0, %5, %11 offset:2048\n\t"
            "global_load_async_to_lds_b128 %0, %6, %11 offset:2560\n\t"
            "global_load_async_to_lds_b128 %0, %7, %11 offset:3072\n\t"
            "global_load_async_to_lds_b128 %0, %8, %11 offset:3584\n\t"
            "global_load_async_to_lds_b128 %0, %9, %11 offset:4096\n\t"
            "global_load_async_to_lds_b128 %0, %10, %11 offset:4608"
            :: "v"(ldsLane),
               "v"(g[0]), "v"(g[1]), "v"(g[2]), "v"(g[3]), "v"(g[4]),
               "v"(g[5]), "v"(g[6]), "v"(g[7]), "v"(g[8]), "v"(g[9]),
               "s"(embed)
            : "memory");
    };

    int h = gwave;
    if (h < totalHalves) issue_half(h, la0);   // prologue fill of buffer 0
    int cur = 0;

    for (; h < totalHalves; h += nwaves) {
        const int  hn      = h + nwaves;
        const bool hasNext = hn < totalHalves;

        if (hasNext) {
            // WAR: prior iteration's LDS reads of this buffer must retire
            // before the async engine overwrites it.
            asm volatile("s_wait_dscnt 0x0" ::: "memory");
            issue_half(hn, cur ? la0 : la1);
            // 10 loads outstanding for the next tile; async loads retire in
            // order per wave, so asynccnt<=10 => current tile complete.
            asm volatile("s_wait_asynccnt 0xA" ::: "memory");
        } else {
            asm volatile("s_wait_asynccnt 0x0" ::: "memory");
        }

        const unsigned char* pc = cur ? pbuf1 : pbuf0;
        v4f acc = (v4f){0.f, 0.f, 0.f, 0.f};
#pragma unroll
        for (int k = 0; k < KNBR; ++k) {
            const v4f v = *(const v4f*)(pc + k * HALF_BYTES + laneByte);
            acc += v;                       // ds_load_b128 + packed adds
        }

        const v4f r = acc * (1.0f / (float)KNBR);

        const int node = h >> 1;
        float* outHalf = out + (size_t)node * DIM + (size_t)(h & 1) * (DIM / 2);
        // Non-temporal: output is write-once; keep it from evicting the
        // L2-resident embed matrix (th:TH_STORE_NT).
        __builtin_nontemporal_store(r, (v4f*)(outHalf + lane * 4));

        cur ^= 1;
    }
}

extern "C" void kernel_launch(void* const* d_in, const int* in_sizes, int n_in,
                              void* d_out, int out_size, void* d_ws, size_t ws_size,
                              hipStream_t stream) {
    (void)n_in; (void)out_size; (void)d_ws; (void)ws_size;

    const int*   nidx  = (const int*)d_in[0];    // neighbor_idx, [N, K]
    const float* embed = (const float*)d_in[1];  // embed_matrix, [M, D]
    float*       outp  = (float*)d_out;          // [N, D]

    const int n_nodes     = in_sizes[0] / KNBR;
    const int totalHalves = n_nodes * 2;

    int blocks = (totalHalves + WAVES_PER_BLK - 1) / WAVES_PER_BLK;
    if (blocks > 1024) blocks = 1024;            // grid-stride; ~24 tiles/wave pipeline
    if (blocks < 1) blocks = 1;

    const size_t shbytes = (size_t)WAVES_PER_BLK * 2u * BUF_BYTES; // 80 KB / block

    mean_agg_async_kernel<<<blocks, WAVES_PER_BLK * 32, shbytes, stream>>>(
        nidx, embed, outp, n_nodes);
}